// GCNPoolBlockLayer_53532472378051
// MI455X (gfx1250) — compile-verified
//
#include <hip/hip_runtime.h>
#include <math.h>

#define Hdim 128
#define Bg   128

typedef float v2f __attribute__((ext_vector_type(2)));
typedef float v8f __attribute__((ext_vector_type(8)));

// ---------------- GEMM: Hout[n,128] = X[n,128] @ W[128,128], fp32 WMMA ----------------
__global__ __launch_bounds__(256) void k_gemm(const float* __restrict__ X,
                                              const float* __restrict__ W,
                                              float* __restrict__ Hout) {
  __shared__ float Wl[Hdim * Hdim];            // 64 KB
  const int tid = threadIdx.x;
  for (int i = tid * 4; i < Hdim * Hdim; i += 256 * 4)
    *(float4*)&Wl[i] = *(const float4*)&W[i];
  __syncthreads();

  const int wave = tid >> 5;
  const int lane = tid & 31;
  const int hs   = lane >> 4;                  // half-wave select (K pair 0/1 vs 2/3)
  const int l16  = lane & 15;
  const long r0  = (long)blockIdx.x * 128 + wave * 16;

  v8f acc[8];
  const v8f zero = {0.f, 0.f, 0.f, 0.f, 0.f, 0.f, 0.f, 0.f};
#pragma unroll
  for (int t = 0; t < 8; ++t) acc[t] = zero;

  const float* xrow = X + (r0 + l16) * Hdim;
  for (int kk = 0; kk < Hdim; kk += 4) {
    const int ka = kk + 2 * hs;
    v2f a;                                     // A 16x4: v0 = K0/K2, v1 = K1/K3
    a.x = xrow[ka];
    a.y = xrow[ka + 1];
#pragma unroll
    for (int t = 0; t < 8; ++t) {
      const int col = t * 16 + l16;
      v2f b;                                   // B 4x16: v0 = rows K0/K2, v1 = K1/K3
      b.x = Wl[ka * Hdim + col];
      b.y = Wl[(ka + 1) * Hdim + col];
      acc[t] = __builtin_amdgcn_wmma_f32_16x16x4_f32(false, a, false, b,
                                                     (short)0, acc[t], false, false);
    }
  }
  const long rb = r0 + hs * 8;                 // C: vgpr j -> row j (lanes 0-15) / 8+j
#pragma unroll
  for (int t = 0; t < 8; ++t)
#pragma unroll
    for (int j = 0; j < 8; ++j)
      Hout[(rb + j) * Hdim + t * 16 + l16] = acc[t][j];
}

// ---------------- small elementwise helpers ----------------
__global__ void k_zero(float* __restrict__ o, int n) {
  int i = blockIdx.x * 256 + threadIdx.x;
  if (i < n) o[i] = 0.f;
}
__global__ void k_fill1(float* __restrict__ d, int n) {
  int i = blockIdx.x * 256 + threadIdx.x;
  if (i < n) d[i] = 1.f;
}
__global__ void k_edges_init(const int* __restrict__ ri, const int* __restrict__ ci,
                             int* __restrict__ er, int* __restrict__ ec,
                             float* __restrict__ mask, int nE) {
  int e = blockIdx.x * 256 + threadIdx.x;
  if (e < nE) { er[e] = ri[e]; ec[e] = ci[e]; mask[e] = 1.f; }
}
__global__ void k_degscatter(const int* __restrict__ ec, const float* __restrict__ mask,
                             float* __restrict__ deg, int nE) {
  int e = blockIdx.x * 256 + threadIdx.x;
  if (e >= nE) return;
  float m = mask[e];
  if (m != 0.f) unsafeAtomicAdd(&deg[ec[e]], m);
}
// agg = h/deg + b   (self-loop dinv^2 = 1/deg, plus bias)
__global__ void k_agginit(const float* __restrict__ h, const float* __restrict__ deg,
                          const float* __restrict__ b, float* __restrict__ agg, int total) {
  int idx = blockIdx.x * 256 + threadIdx.x;
  if (idx >= total) return;
  int i = idx >> 7, f = idx & 127;
  agg[idx] = h[idx] / deg[i] + b[f];
}
__global__ void k_dinv(float* __restrict__ deg, int n) {
  int i = blockIdx.x * 256 + threadIdx.x;
  if (i < n) deg[i] = rsqrtf(deg[i]);
}
// one wave per edge: agg[col] += h[row] * dinv[row]*dinv[col]*mask
__global__ __launch_bounds__(256) void k_edgeagg(const float* __restrict__ h,
                                                 const int* __restrict__ er,
                                                 const int* __restrict__ ec,
                                                 const float* __restrict__ mask,
                                                 const float* __restrict__ dinv,
                                                 float* __restrict__ agg, int nE) {
  int e = blockIdx.x * 8 + (threadIdx.x >> 5);
  if (e >= nE) return;
  int lane = threadIdx.x & 31;
  int r = er[e], c = ec[e];
  float coef = dinv[r] * dinv[c] * mask[e];
  if (coef == 0.f) return;
  float4 hv = *(const float4*)(h + (long)r * Hdim + lane * 4);
  float* dst = agg + (long)c * Hdim + lane * 4;
  unsafeAtomicAdd(dst + 0, hv.x * coef);
  unsafeAtomicAdd(dst + 1, hv.y * coef);
  unsafeAtomicAdd(dst + 2, hv.z * coef);
  unsafeAtomicAdd(dst + 3, hv.w * coef);
}
__global__ void k_relu(float* __restrict__ a, int n) {
  int i = blockIdx.x * 256 + threadIdx.x;
  if (i < n) a[i] = fmaxf(a[i], 0.f);
}
__global__ void k_pnorm(const float* __restrict__ p, float* __restrict__ inv) {
  __shared__ float red[128];
  int t = threadIdx.x;
  float v = p[t];
  red[t] = v * v;
  __syncthreads();
  for (int s = 64; s > 0; s >>= 1) {
    if (t < s) red[t] += red[t + s];
    __syncthreads();
  }
  if (t == 0) inv[0] = rsqrtf(red[0]);
}
// one wave per node: score[i] = dot(x[i], p) / ||p||
__global__ __launch_bounds__(256) void k_score(const float* __restrict__ X,
                                               const float* __restrict__ p,
                                               const float* __restrict__ inv,
                                               float* __restrict__ score, int n) {
  int i = blockIdx.x * 8 + (threadIdx.x >> 5);
  if (i >= n) return;
  int lane = threadIdx.x & 31;
  float4 xv = *(const float4*)(X + (long)i * Hdim + lane * 4);
  float4 pv = *(const float4*)(p + lane * 4);
  float s = xv.x * pv.x + xv.y * pv.y + xv.z * pv.z + xv.w * pv.w;
  for (int off = 16; off > 0; off >>= 1) s += __shfl_down(s, off, 32);
  if (lane == 0) score[i] = s * inv[0];
}
// one block per graph: rank == stable descending-sort position -> newid directly
__global__ __launch_bounds__(256) void k_topk(const float* __restrict__ X,
                                              const float* __restrict__ score,
                                              float* __restrict__ XS,
                                              int* __restrict__ nid,
                                              int n_per, int k) {
  __shared__ float s_lds[1024];
  __shared__ short r_lds[1024];
  int g = blockIdx.x, tid = threadIdx.x;
  const float* sg = score + (long)g * n_per;
  for (int i = tid; i < n_per; i += 256) s_lds[i] = sg[i];
  __syncthreads();
  for (int i = tid; i < n_per; i += 256) {
    float si = s_lds[i];
    int rank = 0;
    for (int j = 0; j < n_per; ++j) {
      float sj = s_lds[j];
      rank += (sj > si) || (sj == si && j < i);
    }
    r_lds[i] = (short)rank;
    nid[(long)g * n_per + i] = (rank < k) ? (g * k + rank) : -1;
  }
  __syncthreads();
  int wave = tid >> 5, lane = tid & 31;
  for (int i = wave; i < n_per; i += 8) {
    int rank = r_lds[i];
    if (rank < k) {
      float t = tanhf(s_lds[i]);
      float4 v = *(const float4*)(X + ((long)g * n_per + i) * Hdim + lane * 4);
      *(float4*)(XS + ((long)g * k + rank) * Hdim + lane * 4) =
          make_float4(v.x * t, v.y * t, v.z * t, v.w * t);
    }
  }
}
__global__ void k_relabel(int* __restrict__ er, int* __restrict__ ec,
                          float* __restrict__ mask, const int* __restrict__ nid, int nE) {
  int e = blockIdx.x * 256 + threadIdx.x;
  if (e >= nE) return;
  int nr = nid[er[e]], nc = nid[ec[e]];
  float m = mask[e];
  if (nr < 0 || nc < 0) m = 0.f;
  er[e] = nr < 0 ? 0 : nr;
  ec[e] = nc < 0 ? 0 : nc;
  mask[e] = m;
}
// one block per graph, thread f = feature: out[g][f] += max_j, out[g][128+f] += mean_j
__global__ __launch_bounds__(128) void k_readout(const float* __restrict__ XS,
                                                 float* __restrict__ out, int k) {
  int g = blockIdx.x, f = threadIdx.x;
  const float* base = XS + (long)g * k * Hdim + f;
  float mx = -INFINITY, sm = 0.f;
  for (int j = 0; j < k; ++j) {
    float v = base[(long)j * Hdim];
    mx = fmaxf(mx, v);
    sm += v;
  }
  out[g * 256 + f] += mx;
  out[g * 256 + 128 + f] += sm / (float)k;
}

// ---------------- host side ----------------
static void run_layer(const float* x, float* hbuf, float* aggbuf, float* xs,
                      const float* W, const float* b, const float* p,
                      int n_per, int k, int nE,
                      int* er, int* ec, float* mask, float* deg,
                      float* score, int* nid, float* inv,
                      float* out, bool relabel, hipStream_t s) {
  const int n = Bg * n_per;
  k_gemm<<<n / 128, 256, 0, s>>>(x, W, hbuf);
  k_fill1<<<(n + 255) / 256, 256, 0, s>>>(deg, n);
  k_degscatter<<<(nE + 255) / 256, 256, 0, s>>>(ec, mask, deg, nE);
  k_agginit<<<(n * Hdim + 255) / 256, 256, 0, s>>>(hbuf, deg, b, aggbuf, n * Hdim);
  k_dinv<<<(n + 255) / 256, 256, 0, s>>>(deg, n);
  k_edgeagg<<<(nE + 7) / 8, 256, 0, s>>>(hbuf, er, ec, mask, deg, aggbuf, nE);
  k_relu<<<(n * Hdim + 255) / 256, 256, 0, s>>>(aggbuf, n * Hdim);
  k_pnorm<<<1, 128, 0, s>>>(p, inv);
  k_score<<<(n + 7) / 8, 256, 0, s>>>(aggbuf, p, inv, score, n);
  k_topk<<<Bg, 256, 0, s>>>(aggbuf, score, xs, nid, n_per, k);
  if (relabel) k_relabel<<<(nE + 255) / 256, 256, 0, s>>>(er, ec, mask, nid, nE);
  k_readout<<<Bg, 128, 0, s>>>(xs, out, k);
}

extern "C" void kernel_launch(void* const* d_in, const int* in_sizes, int n_in,
                              void* d_out, int out_size, void* d_ws, size_t ws_size,
                              hipStream_t stream) {
  const float* x0 = (const float*)d_in[0];
  const int* row_in = (const int*)d_in[1];
  const int* col_in = (const int*)d_in[2];
  const float* W1 = (const float*)d_in[3];
  const float* b1 = (const float*)d_in[4];
  const float* p1 = (const float*)d_in[5];
  const float* W2 = (const float*)d_in[6];
  const float* b2 = (const float*)d_in[7];
  const float* p2 = (const float*)d_in[8];
  const float* W3 = (const float*)d_in[9];
  const float* b3 = (const float*)d_in[10];
  const float* p3 = (const float*)d_in[11];
  float* out = (float*)d_out;

  const int n1 = in_sizes[0] / Hdim;     // 131072
  const int nE = in_sizes[1];            // 1048576
  const size_t BUF = (size_t)n1 * Hdim;  // 16.7M floats

  float* buf0 = (float*)d_ws;
  float* buf1 = buf0 + BUF;
  int* er = (int*)(buf1 + BUF);
  int* ec = er + nE;
  float* mask = (float*)(ec + nE);
  float* deg = mask + nE;
  float* score = deg + n1;
  int* nid = (int*)(score + n1);
  float* inv = (float*)(nid + n1);

  k_zero<<<(out_size + 255) / 256, 256, 0, stream>>>(out, out_size);
  k_edges_init<<<(nE + 255) / 256, 256, 0, stream>>>(row_in, col_in, er, ec, mask, nE);

  // layer 1: x=d_in, h=buf0, agg/out=buf1, xs=buf0
  run_layer(x0, buf0, buf1, buf0, W1, b1, p1, 1024, 512, nE,
            er, ec, mask, deg, score, nid, inv, out, true, stream);
  // layer 2: x=buf0, h=buf1, agg/out=buf0, xs=buf1
  run_layer(buf0, buf1, buf0, buf1, W2, b2, p2, 512, 256, nE,
            er, ec, mask, deg, score, nid, inv, out, true, stream);
  // layer 3: x=buf1, h=buf0, agg/out=buf1, xs=buf0  (no relabel needed)
  run_layer(buf1, buf0, buf1, buf0, W3, b3, p3, 256, 128, nE,
            er, ec, mask, deg, score, nid, inv, out, false, stream);
}